// GliomaGene2VecModel_11785390260745
// MI455X (gfx1250) — compile-verified
//
#include <hip/hip_runtime.h>
#include <math.h>

typedef __attribute__((ext_vector_type(2))) float v2f;
typedef __attribute__((ext_vector_type(8))) float v8f;

#define DIM    128
#define NEG_K  5
#define WAVES_PER_BLOCK 8

__device__ __forceinline__ float log_sigmoid_f(float x) {
    // numerically stable: min(x,0) - log1p(exp(-|x|))
    return fminf(x, 0.0f) - log1pf(expf(-fabsf(x)));
}

__global__ void __launch_bounds__(32 * WAVES_PER_BLOCK)
sgns_wmma_loss_kernel(const float* __restrict__ W_in,
                      const float* __restrict__ W_ctx,
                      const int*   __restrict__ input_word,
                      const int*   __restrict__ context_word,
                      const int*   __restrict__ neg_idx,
                      float*       __restrict__ block_partials,
                      int batch, int ntiles)
{
    __shared__ float warp_part[WAVES_PER_BLOCK];

    const int lane = threadIdx.x & 31;
    const int wave = threadIdx.x >> 5;
    const int tile = blockIdx.x * WAVES_PER_BLOCK + wave;

    float lane_sum = 0.0f;

    if (tile < ntiles) {   // wave-uniform branch: EXEC all-ones inside (WMMA req.)
        const int r    = lane & 15;   // sample row within tile (A row / B col)
        const int half = lane >> 4;   // K-pair selector within 4-wide K step
        int b = tile * 16 + r;
        if (b >= batch) b = batch - 1;   // clamp (tail safety; contribution masked)

        const int iw = input_word[b];
        const int cw = context_word[b];
        const float* pA = W_in  + (size_t)iw * DIM + 2 * half;
        const float* pB = W_ctx + (size_t)cw * DIM + 2 * half;
        const float* pN[NEG_K];
#pragma unroll
        for (int k = 0; k < NEG_K; ++k) {
            const int ni = neg_idx[b * NEG_K + k];
            pN[k] = W_in + (size_t)ni * DIM + 2 * half;
        }

        v8f cpos = {};
        v8f cneg[NEG_K] = {};

        // D = A (16x128 input rows) x B^T, accumulated over 32 K-steps of 4.
        // Per step each lane supplies float2 {row[k0], row[k0+1]} -> VGPR pair.
#pragma unroll 8
        for (int i = 0; i < DIM / 4; ++i) {
            const v2f a  = *(const v2f*)(pA + 4 * i);
            const v2f bb = *(const v2f*)(pB + 4 * i);
            cpos = __builtin_amdgcn_wmma_f32_16x16x4_f32(
                       false, a, false, bb, (short)0, cpos, false, false);
#pragma unroll
            for (int k = 0; k < NEG_K; ++k) {
                const v2f nn = *(const v2f*)(pN[k] + 4 * i);
                cneg[k] = __builtin_amdgcn_wmma_f32_16x16x4_f32(
                              false, a, false, nn, (short)0, cneg[k], false, false);
            }
        }

        // Diagonal extraction (C/D layout): sample m<8 -> lane m, vgpr m;
        // sample m>=8 -> lane 24+(m-8), vgpr m-8. All 6 scores for a sample
        // land in the same lane.
        const bool lo = (lane < 8);
        const bool hi = (lane >= 24);
        const int  e  = lo ? lane : (lane - 24);

        float pos = cpos[0];
        float ng[NEG_K];
#pragma unroll
        for (int k = 0; k < NEG_K; ++k) ng[k] = cneg[k][0];
#pragma unroll
        for (int j = 1; j < 8; ++j) {
            pos = (e == j) ? cpos[j] : pos;
#pragma unroll
            for (int k = 0; k < NEG_K; ++k)
                ng[k] = (e == j) ? cneg[k][j] : ng[k];
        }

        // per-sample loss: log_sigmoid(pos) + sum_k log_sigmoid(-dot_k)
        float loss = log_sigmoid_f(pos);
#pragma unroll
        for (int k = 0; k < NEG_K; ++k) loss += log_sigmoid_f(-ng[k]);

        const int sample = lo ? lane : (lane - 16);          // 0..15 for valid lanes
        const bool valid = (lo || hi) && (tile * 16 + sample < batch);
        lane_sum = valid ? loss : 0.0f;
    }

    // deterministic wave32 reduction
#pragma unroll
    for (int off = 16; off > 0; off >>= 1)
        lane_sum += __shfl_xor(lane_sum, off, 32);

    if (lane == 0) warp_part[wave] = lane_sum;
    __syncthreads();
    if (threadIdx.x == 0) {
        float s = 0.0f;
        for (int w = 0; w < WAVES_PER_BLOCK; ++w) s += warp_part[w];
        block_partials[blockIdx.x] = s;
    }
}

__global__ void __launch_bounds__(512)
sgns_reduce_kernel(const float* __restrict__ parts,
                   float* __restrict__ out, int n, float inv_batch)
{
    __shared__ float s[512];
    const int t = threadIdx.x;
    float v = 0.0f;
    for (int i = t; i < n; i += 512) v += parts[i];   // fixed order per slot
    s[t] = v;
    __syncthreads();
    for (int off = 256; off > 0; off >>= 1) {
        if (t < off) s[t] += s[t + off];
        __syncthreads();
    }
    if (t == 0) out[0] = -s[0] * inv_batch;   // total = -mean(...)
}

extern "C" void kernel_launch(void* const* d_in, const int* in_sizes, int n_in,
                              void* d_out, int out_size, void* d_ws, size_t ws_size,
                              hipStream_t stream) {
    (void)n_in; (void)out_size; (void)ws_size;

    const float* W_in         = (const float*)d_in[0];
    const float* W_ctx        = (const float*)d_in[1];
    const int*   input_word   = (const int*)d_in[2];
    const int*   context_word = (const int*)d_in[3];
    const int*   neg_idx      = (const int*)d_in[4];

    const int batch  = in_sizes[2];                 // 65536
    const int ntiles = (batch + 15) / 16;           // 4096
    const int blocks = (ntiles + WAVES_PER_BLOCK - 1) / WAVES_PER_BLOCK;  // 512

    float* partials = (float*)d_ws;                 // blocks floats of scratch

    sgns_wmma_loss_kernel<<<blocks, 32 * WAVES_PER_BLOCK, 0, stream>>>(
        W_in, W_ctx, input_word, context_word, neg_idx,
        partials, batch, ntiles);

    sgns_reduce_kernel<<<1, 512, 0, stream>>>(
        partials, (float*)d_out, blocks, 1.0f / (float)batch);
}